// SpGraphAttentionLayer_44504451121559
// MI455X (gfx1250) — compile-verified
//
#include <hip/hip_runtime.h>
#include <hip/hip_bf16.h>
#include <math.h>

#define ALPHA 0.2f

typedef __attribute__((ext_vector_type(2))) float v2f;
typedef __attribute__((ext_vector_type(8))) float v8f;

// ---------------------------------------------------------------------------
// Zero-fill helper (hp accumulator + rowsum must start at 0 every call)
// ---------------------------------------------------------------------------
__global__ void zero_kernel(float* __restrict__ p, long n) {
    long t = (long)blockIdx.x * blockDim.x + threadIdx.x;
    if (t < n) p[t] = 0.0f;
}

// ---------------------------------------------------------------------------
// h = x @ W  via V_WMMA_F32_16X16X4_F32 (fp32 matrix core path).
// One wave computes one 16x16 tile of h; 8 waves/block cover a full
// 16-row x 128-col strip. K=128 -> 32 WMMA issues, fully unrolled.
//
// Layouts (ISA 7.12.2, 32-bit operands, wave32):
//   A 16x4: lane l (m=l&15, g=l>>4): v0=A[m][k+2g], v1=A[m][k+2g+1]
//   B 4x16: lane l (n=l&15, g=l>>4): v0=B[k+2g][n], v1=B[k+2g+1][n]
//   C 16x16: VGPR j: lanes0-15 -> M=j, lanes16-31 -> M=j+8, N=l&15
// ---------------------------------------------------------------------------
__global__ __launch_bounds__(256) void gemm_h_kernel(
    const float* __restrict__ x, const float* __restrict__ W,
    float* __restrict__ h, int N) {
    const int lane = threadIdx.x & 31;
    const int wave = threadIdx.x >> 5;          // column tile 0..7
    const int row0 = blockIdx.x * 16;           // N % 16 == 0 (50000 = 3125*16)
    const int col0 = wave * 16;
    const int m = lane & 15;
    const int g = lane >> 4;

    const float* __restrict__ xrow = x + (long)(row0 + m) * 128;
    const float* __restrict__ wcol = W + col0 + m;   // m doubles as n for B

    v8f c = {};
#pragma unroll
    for (int k = 0; k < 128; k += 4) {
        v2f a, b;
        a.x = xrow[k + 2 * g + 0];
        a.y = xrow[k + 2 * g + 1];
        b.x = wcol[(long)(k + 2 * g + 0) * 128];
        b.y = wcol[(long)(k + 2 * g + 1) * 128];
        // 8 args: (neg_a, A, neg_b, B, c_mod, C, reuse_a, reuse_b)
        c = __builtin_amdgcn_wmma_f32_16x16x4_f32(
            false, a, false, b, (short)0, c, false, false);
    }
#pragma unroll
    for (int j = 0; j < 8; ++j) {
        int mm = j + 8 * g;
        h[(long)(row0 + mm) * 128 + col0 + m] = c[j];
    }
}

// ---------------------------------------------------------------------------
// Per-node attention scalars: s_src[i] = h[i].a_src, s_dst[i] = h[i].a_dst
// One wave32 per node, shuffle reduction.
// ---------------------------------------------------------------------------
__global__ __launch_bounds__(256) void node_score_kernel(
    const float* __restrict__ h, const float* __restrict__ a,
    float* __restrict__ s_src, float* __restrict__ s_dst, int N) {
    const int lane = threadIdx.x & 31;
    const int node = blockIdx.x * (blockDim.x >> 5) + (threadIdx.x >> 5);
    if (node >= N) return;
    const float* __restrict__ hr = h + (long)node * 128;
    float ss = 0.0f, sd = 0.0f;
#pragma unroll
    for (int j = lane; j < 128; j += 32) {
        float v = hr[j];
        ss += v * a[j];
        sd += v * a[128 + j];
    }
#pragma unroll
    for (int off = 16; off > 0; off >>= 1) {
        ss += __shfl_down(ss, off, 32);
        sd += __shfl_down(sd, off, 32);
    }
    if (lane == 0) {
        s_src[node] = ss;
        s_dst[node] = sd;
    }
}

// ---------------------------------------------------------------------------
// Per-edge: e = exp(-leakyrelu(s_src[src]+s_dst[dst])) * adj; rowsum[src] += e
// ---------------------------------------------------------------------------
__global__ __launch_bounds__(256) void edge_score_kernel(
    const int* __restrict__ src, const int* __restrict__ dst,
    const float* __restrict__ adj,
    const float* __restrict__ s_src, const float* __restrict__ s_dst,
    float* __restrict__ e_edge, float* __restrict__ rowsum, int E) {
    int e = blockIdx.x * blockDim.x + threadIdx.x;
    if (e >= E) return;
    int s = src[e];
    float sc = s_src[s] + s_dst[dst[e]];
    float lr = sc > 0.0f ? sc : ALPHA * sc;
    float ee = expf(-lr) * adj[e];
    e_edge[e] = ee;
    atomicAdd(&rowsum[s], ee);
}

// ---------------------------------------------------------------------------
// SpMM scatter: hp[src] += e * h[dst].  One wave32 per edge; each lane
// handles a float4 slice (128 floats / 32 lanes). Atomics hit the 192MB L2
// since hp is only 25.6MB.
// ---------------------------------------------------------------------------
__global__ __launch_bounds__(256) void edge_scatter_kernel(
    const int* __restrict__ src, const int* __restrict__ dst,
    const float* __restrict__ e_edge, const float* __restrict__ h,
    float* __restrict__ hp, int E) {
    const int lane = threadIdx.x & 31;
    const int e = blockIdx.x * (blockDim.x >> 5) + (threadIdx.x >> 5);
    if (e >= E) return;
    const float w = e_edge[e];
    const int s = src[e];
    const int d = dst[e];
    const float4 v = ((const float4*)(h + (long)d * 128))[lane];
    float* __restrict__ o = hp + (long)s * 128 + lane * 4;
    atomicAdd(o + 0, w * v.x);
    atomicAdd(o + 1, w * v.y);
    atomicAdd(o + 2, w * v.z);
    atomicAdd(o + 3, w * v.w);
}

// ---------------------------------------------------------------------------
// Finalize: zero-row fallback to h, divide by rowsum, ELU.
// ---------------------------------------------------------------------------
__global__ __launch_bounds__(256) void finalize_kernel(
    const float* __restrict__ h, const float* __restrict__ hp,
    const float* __restrict__ rowsum, float* __restrict__ out, long total) {
    long t = (long)blockIdx.x * blockDim.x + threadIdx.x;
    if (t >= total) return;
    int node = (int)(t >> 7);
    float r = rowsum[node];
    float v = (r == 0.0f) ? h[t] : hp[t] / r;
    out[t] = v > 0.0f ? v : expm1f(v);
}

// ---------------------------------------------------------------------------
// Host launcher
// ---------------------------------------------------------------------------
extern "C" void kernel_launch(void* const* d_in, const int* in_sizes, int n_in,
                              void* d_out, int out_size, void* d_ws, size_t ws_size,
                              hipStream_t stream) {
    const float* x        = (const float*)d_in[0];   // N x 128
    const float* W        = (const float*)d_in[1];   // 128 x 128
    const float* a        = (const float*)d_in[2];   // 1 x 256
    const float* adj_vals = (const float*)d_in[3];   // E
    const int*   eidx     = (const int*)d_in[4];     // 2 x E

    const int N = in_sizes[0] / 128;
    const int E = in_sizes[3];
    const int* src = eidx;
    const int* dst = eidx + E;

    // Workspace layout (floats)
    float* ws     = (float*)d_ws;
    float* h      = ws;                     // N*128
    float* hp     = h  + (long)N * 128;     // N*128
    float* e_edge = hp + (long)N * 128;     // E
    float* s_src  = e_edge + E;             // N
    float* s_dst  = s_src + N;              // N
    float* rowsum = s_dst + N;              // N

    const long NH = (long)N * 128;

    // 1) zero accumulators
    zero_kernel<<<(int)((NH + 255) / 256), 256, 0, stream>>>(hp, NH);
    zero_kernel<<<(N + 255) / 256, 256, 0, stream>>>(rowsum, N);

    // 2) h = x @ W via fp32 WMMA (N multiple of 16)
    gemm_h_kernel<<<N / 16, 256, 0, stream>>>(x, W, h, N);

    // 3) per-node score scalars
    node_score_kernel<<<(N + 7) / 8, 256, 0, stream>>>(h, a, s_src, s_dst, N);

    // 4) per-edge attention weights + rowsum
    edge_score_kernel<<<(E + 255) / 256, 256, 0, stream>>>(
        src, dst, adj_vals, s_src, s_dst, e_edge, rowsum, E);

    // 5) SpMM scatter (wave32 per edge)
    edge_scatter_kernel<<<(E + 7) / 8, 256, 0, stream>>>(
        src, dst, e_edge, h, hp, E);

    // 6) normalize + ELU into d_out
    finalize_kernel<<<(int)((NH + 255) / 256), 256, 0, stream>>>(
        h, hp, rowsum, (float*)d_out, NH);
}